// SoftSkeletonize_5755256176782
// MI455X (gfx1250) — compile-verified
//
#include <hip/hip_runtime.h>
#include <math.h>

#define DIM   192
#define VOL   (DIM*DIM*DIM)      // 7077888
#define NTOT  (2*VOL)            // 14155776
#define NITER 40

// ---------------- CDNA5 async global->LDS support ----------------
#if defined(__has_builtin)
# if __has_builtin(__builtin_amdgcn_global_load_async_to_lds_b32)
#  define HAVE_ASYNC_B32 1
# endif
# if __has_builtin(__builtin_amdgcn_global_load_async_to_lds_b128)
#  define HAVE_ASYNC_B128 1
# endif
# if __has_builtin(__builtin_amdgcn_s_wait_asynccnt)
#  define HAVE_WAIT_ASYNC 1
# endif
#endif

typedef int v4i __attribute__((ext_vector_type(4)));
typedef __attribute__((address_space(1))) int   gas_int;
typedef __attribute__((address_space(3))) int   las_int;
typedef __attribute__((address_space(1))) float gas_float;
typedef __attribute__((address_space(3))) float las_float;
typedef __attribute__((address_space(1))) v4i   gas_v4i;
typedef __attribute__((address_space(3))) v4i   las_v4i;

__device__ __forceinline__ void async_wait_all() {
#if defined(HAVE_WAIT_ASYNC)
  __builtin_amdgcn_s_wait_asynccnt(0);
#else
  asm volatile("s_wait_asynccnt 0" ::: "memory");
#endif
}

__device__ __forceinline__ void async_ld_b32(const float* __restrict__ gp,
                                             float* lp) {
#if defined(HAVE_ASYNC_B32)
  float* gnc = const_cast<float*>(gp);
  gas_float* g1 = (gas_float*)gnc;
  gas_int*   g2 = (gas_int*)g1;
  las_float* l1 = (las_float*)lp;
  las_int*   l2 = (las_int*)l1;
  __builtin_amdgcn_global_load_async_to_lds_b32(g2, l2, 0, 0);
#else
  *lp = *gp;
#endif
}

// 16 bytes, both sides 16B-aligned.
__device__ __forceinline__ void async_ld_b128(const float* __restrict__ gp,
                                              float* lp) {
#if defined(HAVE_ASYNC_B128)
  float* gnc = const_cast<float*>(gp);
  gas_float* g1 = (gas_float*)gnc;
  gas_v4i*   g2 = (gas_v4i*)g1;
  las_float* l1 = (las_float*)lp;
  las_v4i*   l2 = (las_v4i*)l1;
  __builtin_amdgcn_global_load_async_to_lds_b128(g2, l2, 0, 0);
#elif defined(HAVE_ASYNC_B32)
  async_ld_b32(gp + 0, lp + 0);
  async_ld_b32(gp + 1, lp + 1);
  async_ld_b32(gp + 2, lp + 2);
  async_ld_b32(gp + 3, lp + 3);
#else
  *(float4*)lp = *(const float4*)gp;
#endif
}

__device__ __forceinline__ int clampi(int v) {
  return v < 0 ? 0 : (v > DIM - 1 ? DIM - 1 : v);
}
__device__ __forceinline__ float min7(const float* P, int a, int sx, int sxy) {
  return fminf(P[a],
         fminf(fminf(P[a - 1],   P[a + 1]),
         fminf(fminf(P[a - sx],  P[a + sx]),
               fminf(P[a - sxy], P[a + sxy]))));
}

struct F8 { float s[8]; };
__device__ __forceinline__ F8 ld8(const float* p) {   // p 16B aligned
  F8 r;
  float4 a = *(const float4*)p;
  float4 b = *(const float4*)(p + 4);
  r.s[0] = a.x; r.s[1] = a.y; r.s[2] = a.z; r.s[3] = a.w;
  r.s[4] = b.x; r.s[5] = b.y; r.s[6] = b.z; r.s[7] = b.w;
  return r;
}
__device__ __forceinline__ void st4(float* p, float a, float b, float c, float d) {
  *(float4*)p = make_float4(a, b, c, d);
}

// Common tile shape
#define TX 32
#define TY 8
#define TZ 4

// ======================================================================
// Init kernel (runs once): skel = relu(img - dilate27(erode7(img)))
// ======================================================================
#define IAX (TX+4)
#define IAY (TY+4)
#define IAZ (TZ+4)
#define INA (IAX*IAY*IAZ)         // 3456
#define IEX (TX+2)
#define IEY (TY+2)
#define IEZ (TZ+2)
#define INE (IEX*IEY*IEZ)         // 2040

__global__ __launch_bounds__(256)
void open_skel_kernel(const float* __restrict__ im, float* __restrict__ skel) {
  __shared__ float SM[INA + INE];
  float* A = SM;
  float* E = SM + INA;

  const int tid = threadIdx.x;
  const int x0 = blockIdx.x * TX;
  const int y0 = blockIdx.y * TY;
  const int zz0 = blockIdx.z * TZ;
  const int b  = zz0 / DIM;
  const int z0 = zz0 - b * DIM;
  const size_t bbase = (size_t)b * VOL;

  #pragma unroll
  for (int k = 0; k < (INA + 255) / 256; ++k) {
    int i = tid + k * 256;
    if (i > INA - 1) i = INA - 1;
    int az = i / (IAX * IAY);
    int r  = i - az * (IAX * IAY);
    int ay = r / IAX;
    int ax = r - ay * IAX;
    int gx = clampi(x0 + ax - 2);
    int gy = clampi(y0 + ay - 2);
    int gz = clampi(z0 + az - 2);
    async_ld_b32(im + (bbase + ((size_t)gz * DIM + gy) * DIM + gx), &A[i]);
  }
  async_wait_all();
  __syncthreads();

  #pragma unroll
  for (int k = 0; k < (INE + 255) / 256; ++k) {
    int i = tid + k * 256;
    if (i > INE - 1) i = INE - 1;
    int ez = i / (IEX * IEY);
    int r  = i - ez * (IEX * IEY);
    int ey = r / IEX;
    int ex = r - ey * IEX;
    int a = ((ez + 1) * IAY + (ey + 1)) * IAX + (ex + 1);
    E[i] = min7(A, a, IAX, IAX * IAY);
  }
  __syncthreads();

  #pragma unroll
  for (int k = 0; k < (TX * TY * TZ) / 256; ++k) {
    int i = tid + k * 256;
    int cz = i >> 8;
    int r  = i & 255;
    int cy = r >> 5;
    int cx = r & 31;
    int e0 = (cz * IEY + cy) * IEX + cx;
    float mx = E[e0];
    #pragma unroll
    for (int dz = 0; dz < 3; ++dz) {
      int pz = e0 + dz * IEX * IEY;
      #pragma unroll
      for (int dy = 0; dy < 3; ++dy) {
        int bb = pz + dy * IEX;
        mx = fmaxf(mx, fmaxf(fmaxf(E[bb], E[bb + 1]), E[bb + 2]));
      }
    }
    float imc   = A[((cz + 2) * IAY + (cy + 2)) * IAX + (cx + 2)];
    float delta = fmaxf(imc - mx, 0.0f);
    size_t g = bbase + ((size_t)(z0 + cz) * DIM + (y0 + cy)) * DIM + (x0 + cx);
    skel[g] = delta;   // skel==0 case of the accumulate rule
  }
}

// ======================================================================
// Fused iteration kernel. All LDS buffers use x-stride 40 (160B rows)
// so every vector access is 16B aligned.
//   A: im tile, window gx = x0 + ax - 4, ax in [0,40), rows 14 x 10
//   B: e1 = erode(im), bx in [0,36), rows 12 x 8      (B = im_next)
//   C: e2 = erode(e1), cx in [0,34) (+pad), rows 10 x 6, aliases A
// ======================================================================
#define XS   40                   // LDS x-stride (floats)
#define AROWS (14*10)             // 140
#define ANF  (XS*14*10)           // 5600
#define BNF  (XS*12*8)            // 3840
#define ASLAB (XS*14)             // 560
#define BSLAB (XS*12)             // 480
#define CSLAB (XS*10)             // 400

__global__ __launch_bounds__(256)
void step_kernel(const float* __restrict__ im, float* __restrict__ imnext,
                 float* __restrict__ skel) {
  __shared__ __align__(16) float SM[ANF + BNF];
  float* A = SM;
  float* B = SM + ANF;
  float* C = SM;                  // aliases A (dead after phase 2)

  const int tid = threadIdx.x;
  const int x0 = blockIdx.x * TX;
  const int y0 = blockIdx.y * TY;
  const int zz0 = blockIdx.z * TZ;
  const int b  = zz0 / DIM;
  const int z0 = zz0 - b * DIM;
  const size_t bbase = (size_t)b * VOL;

  // ---- P1: async-stage A rows (gy/gz clamped per row) ----
  if (x0 != 0 && x0 != DIM - TX) {
    // interior in x: whole window [x0-4, x0+36) is in range -> vector rows
    #pragma unroll
    for (int p = 0; p < 9; ++p) {                 // 16 rows x 16 lanes
      int row = p * 16 + (tid >> 4);
      if (row > AROWS - 1) row = AROWS - 1;
      int sub = tid & 15;
      if (sub > 9) sub = 9;                       // dup loads benign
      int az = row / 14;
      int ay = row - az * 14;
      int gy = clampi(y0 + ay - 3);
      int gz = clampi(z0 + az - 3);
      const float* gp = im + (bbase + ((size_t)gz * DIM + gy) * DIM
                              + (x0 - 4 + sub * 4));
      async_ld_b128(gp, &A[row * XS + sub * 4]);
    }
  } else {
    // x edge: scalar rows with clamped gx
    #pragma unroll
    for (int p = 0; p < 18; ++p) {                // 8 rows x 32 lanes
      int row = p * 8 + (tid >> 5);
      if (row > AROWS - 1) row = AROWS - 1;
      int lane = tid & 31;
      int az = row / 14;
      int ay = row - az * 14;
      int gy = clampi(y0 + ay - 3);
      int gz = clampi(z0 + az - 3);
      size_t rb = bbase + ((size_t)gz * DIM + gy) * DIM;
      int ax1 = lane;
      int ax2 = lane < 8 ? lane + 32 : 39;        // dup loads benign
      async_ld_b32(im + rb + clampi(x0 + ax1 - 4), &A[row * XS + ax1]);
      async_ld_b32(im + rb + clampi(x0 + ax2 - 4), &A[row * XS + ax2]);
    }
  }
  async_wait_all();
  __syncthreads();

  // ---- P2: B = erode7(A), 36x12x8 outputs as 864 float4 groups ----
  #pragma unroll
  for (int p = 0; p < 4; ++p) {
    int g = tid + p * 256;
    if (g > 863) g = 863;
    int row = g / 9;                 // 0..95  (bz*12+by)
    int sub = g - row * 9;
    int bz = row / 12;
    int by = row - bz * 12;
    int bx = sub * 4;                // output x base, center ax = bx+2+j
    int ac = ((bz + 1) * 14 + (by + 1)) * XS;
    F8 xc = ld8(&A[ac + bx]);
    F8 ym = ld8(&A[ac - XS    + bx]);
    F8 yp = ld8(&A[ac + XS    + bx]);
    F8 zm = ld8(&A[ac - ASLAB + bx]);
    F8 zp = ld8(&A[ac + ASLAB + bx]);
    float o[4];
    #pragma unroll
    for (int j = 0; j < 4; ++j) {
      o[j] = fminf(xc.s[2 + j],
             fminf(fminf(xc.s[1 + j], xc.s[3 + j]),
             fminf(fminf(ym.s[2 + j], yp.s[2 + j]),
                   fminf(zm.s[2 + j], zp.s[2 + j]))));
    }
    st4(&B[row * XS + bx], o[0], o[1], o[2], o[3]);
  }
  __syncthreads();

  // ---- P3: C = erode7(B), 36(34 used)x10x6 outputs, 540 groups ----
  #pragma unroll
  for (int p = 0; p < 3; ++p) {
    int g = tid + p * 256;
    if (g > 539) g = 539;
    int row = g / 9;                 // 0..59 (cz*10+cy)
    int sub = g - row * 9;
    int cz = row / 10;
    int cy = row - cz * 10;
    int cx = sub * 4;                // center bx = cx+1+j
    int bc = ((cz + 1) * 12 + (cy + 1)) * XS;
    F8 xc = ld8(&B[bc + cx]);
    F8 ym = ld8(&B[bc - XS    + cx]);
    F8 yp = ld8(&B[bc + XS    + cx]);
    F8 zm = ld8(&B[bc - BSLAB + cx]);
    F8 zp = ld8(&B[bc + BSLAB + cx]);
    float o[4];
    #pragma unroll
    for (int j = 0; j < 4; ++j) {
      o[j] = fminf(xc.s[1 + j],
             fminf(fminf(xc.s[0 + j], xc.s[2 + j]),
             fminf(fminf(ym.s[1 + j], yp.s[1 + j]),
                   fminf(zm.s[1 + j], zp.s[1 + j]))));
    }
    st4(&C[row * XS + cx], o[0], o[1], o[2], o[3]);
  }
  __syncthreads();

  // ---- P4: open = dilate27(C); im_next = B center; skel RMW (float4) ----
  {
    int cx0 = (tid & 7) * 4;         // outputs cx0..cx0+3
    int cy  = (tid >> 3) & 7;
    int cz  = tid >> 6;
    // vertical running max over the 9 (dy,dz) rows, 6-wide window
    float m[6];
    {
      F8 r0 = ld8(&C[(cz * 10 + cy) * XS + cx0]);
      #pragma unroll
      for (int i = 0; i < 6; ++i) m[i] = r0.s[i];
    }
    #pragma unroll
    for (int dz = 0; dz < 3; ++dz) {
      #pragma unroll
      for (int dy = 0; dy < 3; ++dy) {
        if (dz == 0 && dy == 0) continue;
        F8 r = ld8(&C[((cz + dz) * 10 + (cy + dy)) * XS + cx0]);
        #pragma unroll
        for (int i = 0; i < 6; ++i) m[i] = fmaxf(m[i], r.s[i]);
      }
    }
    F8 bc = ld8(&B[((cz + 2) * 12 + (cy + 2)) * XS + cx0]);  // im_next vals
    size_t g = bbase + ((size_t)(z0 + cz) * DIM + (y0 + cy)) * DIM + (x0 + cx0);
    float4 sv = *(float4*)&skel[g];
    float sc[4] = {sv.x, sv.y, sv.z, sv.w};
    float imn[4], sk[4];
    #pragma unroll
    for (int j = 0; j < 4; ++j) {
      float open  = fmaxf(fmaxf(m[j], m[j + 1]), m[j + 2]);
      float imc   = bc.s[2 + j];
      float delta = fmaxf(imc - open, 0.0f);
      imn[j] = imc;
      sk[j]  = sc[j] + fmaxf(delta - sc[j] * delta, 0.0f);
    }
    st4(&imnext[g], imn[0], imn[1], imn[2], imn[3]);
    st4(&skel[g],   sk[0],  sk[1],  sk[2],  sk[3]);
  }
}

// ---------------- Host launch ----------------
extern "C" void kernel_launch(void* const* d_in, const int* in_sizes, int n_in,
                              void* d_out, int out_size, void* d_ws, size_t ws_size,
                              hipStream_t stream) {
  (void)in_sizes; (void)n_in; (void)out_size; (void)ws_size;
  const float* img  = (const float*)d_in[0];
  float* skel = (float*)d_out;
  float* wsA  = (float*)d_ws;
  float* wsB  = wsA + (size_t)NTOT;

  dim3 blk(256);
  dim3 gridU(DIM / TX, DIM / TY, (2 * DIM) / TZ);   // 6 x 24 x 96

  // skel = relu(img - open(img))
  open_skel_kernel<<<gridU, blk, 0, stream>>>(img, skel);

  const float* cur = img;
  for (int it = 0; it < NITER; ++it) {
    float* nxt = (it & 1) ? wsB : wsA;
    step_kernel<<<gridU, blk, 0, stream>>>(cur, nxt, skel);
    cur = nxt;
  }
}